// SparseGrid_32177894981983
// MI455X (gfx1250) — compile-verified
//
#include <hip/hip_runtime.h>
#include <stdint.h>

// Grid geometry fixed by the reference module constants.
#define TRES 600
#define XRES 300
#define YRES 300
#define ROW_F 18      // 9 cells * LEVEL_DIM(2) floats per output row
#define PTS 256       // points per block (8 wave32s)

#if defined(__AMDGCN__)
#  if __has_builtin(__builtin_amdgcn_global_load_async_to_lds_b64) && \
      __has_builtin(__builtin_amdgcn_global_store_async_from_lds_b128) && \
      __has_builtin(__builtin_amdgcn_global_store_async_from_lds_b64)
#    define USE_ASYNC_LDS 1
#  else
#    define USE_ASYNC_LDS 0
#    warning "CDNA5 async-LDS builtins not found on this toolchain; using VGPR fallback path"
#  endif
#else
#  define USE_ASYNC_LDS 0
#endif

// Exact parameter vector types per hipcc's diagnostics (round 1):
typedef int avec2 __attribute__((vector_size(2 * sizeof(int))));
typedef int avec4 __attribute__((vector_size(4 * sizeof(int))));

// gfx12+ CPol: TH in bits[2:0] (RT=0, NT=1), scope in bits[4:3] (CU=0).
// Output is a write-once stream; NT keeps the 192MB L2 for embedding reuse.
#define CPOL_RT 0
#define CPOL_NT_STORE 1

__device__ __forceinline__ int clampi(int v, int lo, int hi) {
    return v < lo ? lo : (v > hi ? hi : v);
}

__global__ __launch_bounds__(256)
void sparsegrid_gather_kernel(const float* __restrict__ in,
                              const float* __restrict__ emb,
                              float* __restrict__ out, int n)
{
#if defined(__HIP_DEVICE_COMPILE__)
    __shared__ float sIn[PTS * 3];                       // staged coords (coalesced)
    __shared__ __align__(16) float sOut[PTS * ROW_F];    // 18432 B staging tile

    const int tid  = threadIdx.x;
    const int base = (int)blockIdx.x * PTS;              // n <= ~2M: int32 is plenty
    const int remn = n - base;
    const int pts  = remn < PTS ? remn : PTS;

    // ---- Phase 0: coalesced load of (pts,3) coords into LDS ----
    {
        const int b3  = base * 3;
        const int lim = pts * 3;
        #pragma unroll
        for (int it = 0; it < 3; ++it) {
            const int k = it * PTS + tid;
            if (k < lim) sIn[k] = in[b3 + k];
        }
    }
    __syncthreads();

    // ---- Phase 1: per-point 3x3 gather, memory -> LDS (async, no VGPR data) ----
    if (tid < pts) {
        const float tv = sIn[tid * 3 + 0];
        const float xv = sIn[tid * 3 + 1];
        const float yv = sIn[tid * 3 + 2];
        // mul-then-add (no FMA contraction), truncate, clip: matches reference.
        const int ti = clampi((int)__fadd_rn(__fmul_rn(tv, (float)(TRES - 1)), 0.5f), 0, TRES - 1);
        const int xi = clampi((int)__fadd_rn(__fmul_rn(xv, (float)(XRES - 1)), 0.5f), 0, XRES - 1);
        const int yi = clampi((int)__fadd_rn(__fmul_rn(yv, (float)(YRES - 1)), 0.5f), 0, YRES - 1);
        #pragma unroll
        for (int ix = 0; ix < 3; ++ix) {
            const int xe = clampi(xi + ix - 1, 0, XRES - 1);
            const int rowbase = (ti * XRES + xe) * (YRES * 2);   // max ~107.8M: fits int32
            #pragma unroll
            for (int iy = 0; iy < 3; ++iy) {
                const int ye = clampi(yi + iy - 1, 0, YRES - 1);
                const float* src = emb + (size_t)(unsigned)(rowbase + ye * 2);
                float* dst = &sOut[tid * ROW_F + (ix * 3 + iy) * 2];
#if USE_ASYNC_LDS
                __builtin_amdgcn_global_load_async_to_lds_b64(
                    (avec2*)src, (avec2*)dst, /*offset=*/0, CPOL_RT);
#else
                *(float2*)dst = *(const float2*)src;     // 8B-aligned (even float offsets)
#endif
            }
        }
    }

#if USE_ASYNC_LDS
#  if __has_builtin(__builtin_amdgcn_s_wait_asynccnt)
    __builtin_amdgcn_s_wait_asynccnt(0);
#  else
    asm volatile("s_wait_asynccnt 0" ::: "memory");
#  endif
#endif
    __syncthreads();

    // ---- Phase 2: LDS -> memory, perfectly coalesced 16B/lane chunks ----
    float* gout = out + (size_t)(unsigned)(base * ROW_F);
    if (pts == PTS) {
        // Full block: 256*18 dwords = exactly 1152 aligned float4 chunks.
        #pragma unroll
        for (int it = 0; it < 4; ++it) {
            const int c = it * PTS + tid;
#if USE_ASYNC_LDS
            __builtin_amdgcn_global_store_async_from_lds_b128(
                (avec4*)(gout + c * 4), (avec4*)&sOut[c * 4], 0, CPOL_NT_STORE);
#else
            *(float4*)(gout + c * 4) = *(const float4*)&sOut[c * 4];
#endif
        }
        if (tid < PTS / 2) {                 // remaining 128 chunks
            const int c = 4 * PTS + tid;
#if USE_ASYNC_LDS
            __builtin_amdgcn_global_store_async_from_lds_b128(
                (avec4*)(gout + c * 4), (avec4*)&sOut[c * 4], 0, CPOL_NT_STORE);
#else
            *(float4*)(gout + c * 4) = *(const float4*)&sOut[c * 4];
#endif
        }
    } else {
        const int dtot   = pts * ROW_F;
        const int chunks = dtot >> 2;
        for (int c = tid; c < chunks; c += PTS) {
#if USE_ASYNC_LDS
            __builtin_amdgcn_global_store_async_from_lds_b128(
                (avec4*)(gout + c * 4), (avec4*)&sOut[c * 4], 0, CPOL_NT_STORE);
#else
            *(float4*)(gout + c * 4) = *(const float4*)&sOut[c * 4];
#endif
        }
        if ((dtot & 3) && tid == 0) {        // pts odd -> 2 leftover dwords
            const int c0 = chunks * 4;
#if USE_ASYNC_LDS
            __builtin_amdgcn_global_store_async_from_lds_b64(
                (avec2*)(gout + c0), (avec2*)&sOut[c0], 0, CPOL_NT_STORE);
#else
            gout[c0] = sOut[c0]; gout[c0 + 1] = sOut[c0 + 1];
#endif
        }
    }
    // S_ENDPGM performs an implicit wait-idle, covering outstanding async stores.
#endif // __HIP_DEVICE_COMPILE__
}

extern "C" void kernel_launch(void* const* d_in, const int* in_sizes, int n_in,
                              void* d_out, int out_size, void* d_ws, size_t ws_size,
                              hipStream_t stream) {
    const float* in  = (const float*)d_in[0];   // (N,3) float32
    const float* emb = (const float*)d_in[1];   // (600,300,300,2) float32
    float* out = (float*)d_out;                 // (N,18) float32
    const int n = in_sizes[0] / 3;
    if (n <= 0) return;
    const int blocks = (n + PTS - 1) / PTS;
    sparsegrid_gather_kernel<<<blocks, PTS, 0, stream>>>(in, emb, out, n);
}